// CapsuleLayer_22041772163089
// MI455X (gfx1250) — compile-verified
//
#include <hip/hip_runtime.h>
#include <hip/hip_bf16.h>
#include <math.h>

// ---------------------------------------------------------------------------
// Capsule routing for MI455X (gfx1250, wave32).
//
// Strategy (see analysis): never materialize u_hat (189 MB). Both per-iteration
// contractions are recast as f32 GEMMs executed with V_WMMA_F32_16X16X4_F32,
// with all operands (x, W, G ~ 23 MB) resident in the 192 MB L2.
//   s[(j,d),b]   = sum_{c,k} (cij[c,j]*W[c,j,d,k]) * x[b,k,c]   M=160 N=256 K=9216
//   G[(j,d),(kc)]= sum_b     v[b,j,d] * x[b,k,c]                M=160 N=9216 K=256
//   agr[c,j]     = sum_{d,k} W[c,j,d,k] * G[j,d,k,c]            (tiny reduce)
// ---------------------------------------------------------------------------

typedef float v2f __attribute__((ext_vector_type(2)));
typedef float v8f __attribute__((ext_vector_type(8)));

namespace caps {
constexpr int B = 256, K = 8, C = 1152, J = 10, D = 16, NITER = 4;
constexpr int M = J * D;          // 160 (M-tile mt == j since D == 16)
constexpr int NK = K * C;         // 9216 (x viewed as (B, 9216))
constexpr int NCH = 9;            // c-chunks for s-GEMM parallelism (deterministic)
constexpr int CCH = C / NCH;      // 128
}

using namespace caps;

// --- one-time input transposes (coalesced fragment loads later) -------------
// xt[c][k][b]  <- x[b][k][c]
__global__ void k_transpose_x(const float* __restrict__ x, float* __restrict__ xt) {
  int idx = blockIdx.x * 256 + threadIdx.x;          // C*K*B = 2359296 exact
  int b = idx % B; int t = idx / B; int k = t % K; int c = t / K;
  xt[idx] = x[(b * K + k) * C + c];
}

// Wt[cj][kb][lane][e] <- W[cj][d = lane&15][k = kb*4 + (lane>>4)*2 + e]
// i.e. exactly the V_WMMA A-fragment order for the (j,d) x k layout.
__global__ void k_transpose_w(const float* __restrict__ W, float* __restrict__ Wt) {
  int idx = blockIdx.x * 256 + threadIdx.x;          // C*J*128 = 1474560 exact
  int e = idx & 1; int l = (idx >> 1) & 31; int kb = (idx >> 6) & 1; int cj = idx >> 7;
  int d = l & 15; int k = kb * 4 + (l >> 4) * 2 + e;
  Wt[idx] = W[(cj * D + d) * K + k];
}

__global__ void k_zero_bij(float* __restrict__ bij) {
  int i = blockIdx.x * 256 + threadIdx.x;            // C*J = 11520 exact (45 blocks)
  bij[i] = 0.0f;
}

// --- softmax over J per input capsule c -------------------------------------
__global__ void k_softmax(const float* __restrict__ bij, float* __restrict__ cij) {
  int c = blockIdx.x * 256 + threadIdx.x;
  if (c >= C) return;
  float t[J]; float mx = -INFINITY;
  #pragma unroll
  for (int j = 0; j < J; ++j) { t[j] = bij[c * J + j]; mx = fmaxf(mx, t[j]); }
  float s = 0.0f;
  #pragma unroll
  for (int j = 0; j < J; ++j) { t[j] = expf(t[j] - mx); s += t[j]; }
  float inv = 1.0f / s;
  #pragma unroll
  for (int j = 0; j < J; ++j) cij[c * J + j] = t[j] * inv;
}

// --- s-GEMM: partial[(ch)][(j,d)][b] via WMMA f32 16x16x4 -------------------
// One 16x16 output tile per wave; contraction split into NCH fixed c-chunks.
// 1440 waves = 180 blocks * 8 waves. EXEC all ones (no divergence).
__global__ void __launch_bounds__(256) k_sgemm(const float* __restrict__ Wt,
                                               const float* __restrict__ xt,
                                               const float* __restrict__ cij,
                                               float* __restrict__ part) {
  int wave  = (blockIdx.x * 256 + threadIdx.x) >> 5;  // 0..1439
  int lane  = threadIdx.x & 31;
  int chunk = wave / (M / 16 * B / 16);               // 0..8
  int tile  = wave % (M / 16 * B / 16);               // 0..159
  int mt = tile / (B / 16);                           // j, 0..9
  int nt = tile % (B / 16);                           // 0..15
  int half = lane >> 4, lm = lane & 15;

  v8f acc = {};
  int c0 = chunk * CCH;
  for (int c = c0; c < c0 + CCH; ++c) {
    const float* wp = Wt + (c * J + mt) * 128 + lane * 2;
    const float* xp = xt + c * (K * B) + nt * 16 + lm;
    __builtin_prefetch(wp + J * 128, 0, 1);           // next c, same j
    __builtin_prefetch(xp + K * B, 0, 1);             // next c
    float sc = cij[c * J + mt];                       // wave-uniform scale
    #pragma unroll
    for (int kb = 0; kb < 2; ++kb) {
      v2f a; a.x = sc * wp[kb * 64 + 0]; a.y = sc * wp[kb * 64 + 1];
      int kk = kb * 4 + half * 2;
      v2f b; b.x = xp[(kk + 0) * B];     b.y = xp[(kk + 1) * B];
      acc = __builtin_amdgcn_wmma_f32_16x16x4_f32(false, a, false, b,
                                                  (short)0, acc, false, false);
    }
  }
  float* pp = part + (chunk * M + mt * 16) * B + nt * 16;
  #pragma unroll
  for (int r = 0; r < 8; ++r) pp[(r + half * 8) * B + lm] = acc[r];
}

// --- reduce partials + squash (s += 1e-5 BEFORE magnitudes, per reference) --
__global__ void k_squash(const float* __restrict__ part, float* __restrict__ vout) {
  int tid = blockIdx.x * 256 + threadIdx.x;           // B*J = 2560 exact (10 blocks)
  int b = tid % B; int j = tid / B;
  float s[D]; float mag = 0.0f;
  #pragma unroll
  for (int d = 0; d < D; ++d) {
    float a = 0.0f;
    #pragma unroll
    for (int ch = 0; ch < NCH; ++ch) a += part[(ch * M + j * 16 + d) * B + b];
    a += 1e-5f;
    s[d] = a; mag = fmaf(a, a, mag);
  }
  float scale = (mag / (1.0f + mag)) * (1.0f / sqrtf(mag));
  #pragma unroll
  for (int d = 0; d < D; ++d) vout[b * M + j * 16 + d] = s[d] * scale;
}

// --- G-GEMM: G[(j,d)][(k,c)] = sum_b v[b][(j,d)] * x[b][(k,c)] --------------
// x in its ORIGINAL layout is already (B, 9216) row-major: perfect B-operand.
// 5760 waves = 720 blocks * 8 waves. EXEC all ones.
__global__ void __launch_bounds__(256) k_ggemm(const float* __restrict__ v,
                                               const float* __restrict__ x,
                                               float* __restrict__ G) {
  int wave = (blockIdx.x * 256 + threadIdx.x) >> 5;   // 0..5759
  int lane = threadIdx.x & 31;
  int mt = wave / (NK / 16);                          // j, 0..9
  int nt = wave % (NK / 16);                          // 0..575
  int half = lane >> 4, lm = lane & 15;

  const float* vp = v + mt * 16 + lm;                 // + b*160
  const float* xp = x + nt * 16 + lm;                 // + b*9216
  v8f acc = {};
  for (int kb = 0; kb < B / 4; ++kb) {
    int b0 = kb * 4 + half * 2;
    __builtin_prefetch(xp + (b0 + 4) * NK, 0, 1);
    v2f a; a.x = vp[(b0 + 0) * M];  a.y = vp[(b0 + 1) * M];
    v2f b; b.x = xp[(b0 + 0) * NK]; b.y = xp[(b0 + 1) * NK];
    acc = __builtin_amdgcn_wmma_f32_16x16x4_f32(false, a, false, b,
                                                (short)0, acc, false, false);
  }
  float* gp = G + (mt * 16) * NK + nt * 16;
  #pragma unroll
  for (int r = 0; r < 8; ++r) gp[(r + half * 8) * NK + lm] = acc[r];
}

// --- agreement: b_ij[c,j] += sum_{d,k} W[c,j,d,k] * G[(j,d)][(k,c)] ---------
__global__ void k_agree(const float* __restrict__ W, const float* __restrict__ G,
                        float* __restrict__ bij) {
  int tid = blockIdx.x * 256 + threadIdx.x;           // J*C = 11520 exact (45 blocks)
  int c = tid % C; int j = tid / C;                   // c fastest -> coalesced G reads
  const float* wp = W + (c * J + j) * (D * K);
  const float* gp = G + (j * 16) * NK + c;
  float acc = 0.0f;
  #pragma unroll
  for (int d = 0; d < D; ++d)
    #pragma unroll
    for (int k = 0; k < K; ++k)
      acc = fmaf(wp[d * K + k], gp[d * NK + k * C], acc);
  bij[c * J + j] += acc;
}

extern "C" void kernel_launch(void* const* d_in, const int* in_sizes, int n_in,
                              void* d_out, int out_size, void* d_ws, size_t ws_size,
                              hipStream_t stream) {
  const float* x = (const float*)d_in[0];   // (B, K, C)
  const float* W = (const float*)d_in[1];   // (1, C, J, D, K)
  float* out = (float*)d_out;               // (B, J, D, 1) == b*160 + j*16 + d
  float* ws = (float*)d_ws;

  // workspace layout (floats), total ~5.74M floats = 23 MB
  float* xt   = ws;                         // 2359296
  float* Wt   = xt   + (size_t)C * K * B;   // 1474560
  float* G    = Wt   + (size_t)C * J * 128; // 1474560
  float* part = G    + (size_t)M * NK;      // 368640
  float* bij  = part + (size_t)NCH * M * B; // 11520
  float* cij  = bij  + (size_t)C * J;       // 11520
  float* v    = cij  + (size_t)C * J;       // 40960

  k_transpose_x<<<(C * K * B) / 256, 256, 0, stream>>>(x, xt);
  k_transpose_w<<<(C * J * 128) / 256, 256, 0, stream>>>(W, Wt);
  k_zero_bij<<<(C * J) / 256, 256, 0, stream>>>(bij);

  for (int it = 0; it < NITER; ++it) {
    k_softmax<<<(C + 255) / 256, 256, 0, stream>>>(bij, cij);
    k_sgemm<<<(NCH * M / 16 * B / 16) / 8, 256, 0, stream>>>(Wt, xt, cij, part);
    float* vtgt = (it == NITER - 1) ? out : v;
    k_squash<<<(B * J) / 256, 256, 0, stream>>>(part, vtgt);
    if (it < NITER - 1) {
      k_ggemm<<<(M / 16 * NK / 16) / 8, 256, 0, stream>>>(v, x, G);
      k_agree<<<(C * J) / 256, 256, 0, stream>>>(W, G, bij);
    }
  }
}